// LocalityPreservingSparsification_68436008894843
// MI455X (gfx1250) — compile-verified
//
#include <hip/hip_runtime.h>

#define HIDDEN 128

typedef __attribute__((ext_vector_type(2))) float v2f;
typedef __attribute__((ext_vector_type(8))) float v8f;

// ---------------------------------------------------------------------------
// Kernel 1: per-node stats. One wave32 per node; each lane loads a float4
// (32 lanes * 4 = 128 elements, one coalesced b128 per row). Also serves to
// pull all of z (51.2 MB) into the 192 MB L2 so the edge-pass gathers hit L2.
// Stores invn[n] = 1/max(||z_n||,1e-12) and pgfi[n] = invn[n]*pgf[n].
// ---------------------------------------------------------------------------
__global__ __launch_bounds__(256) void node_stats_kernel(
    const float* __restrict__ z, float* __restrict__ invn,
    float* __restrict__ pgfi, int n_nodes) {
  const int wid  = threadIdx.x >> 5;
  const int lane = threadIdx.x & 31;
  const int node = blockIdx.x * 8 + wid;
  if (node >= n_nodes) return;

  const float4 v = ((const float4*)(z + (size_t)node * HIDDEN))[lane];
  float s  = v.x + v.y + v.z + v.w;
  float ss = v.x * v.x + v.y * v.y + v.z * v.z + v.w * v.w;

#pragma unroll
  for (int off = 16; off >= 1; off >>= 1) {   // wave32 butterfly reduction
    s  += __shfl_xor(s,  off, 32);
    ss += __shfl_xor(ss, off, 32);
  }

  if (lane == 0) {
    float norm = sqrtf(ss);
    float inv  = 1.0f / fmaxf(norm, 1e-12f);
    float mu   = s * (1.0f / HIDDEN);
    float var  = (ss - (float)HIDDEN * mu * mu) * (1.0f / (HIDDEN - 1));
    var        = fmaxf(var, 0.0f);            // unbiased var, guard round-off
    float pgf  = 1.0f + mu + 0.5f * (mu * mu + var);
    invn[node] = inv;
    pgfi[node] = inv * pgf;
  }
}

// ---------------------------------------------------------------------------
// Kernel 2: per-edge cosine * pgf via V_WMMA_F32_16X16X4_F32.
// One wave per 16-edge tile. A = 16 src rows (16xK), B = 16 dst rows as KxN;
// accumulate C(16x16) over K=128 in 32 steps of K=4; diagonal = the 16 dots.
//
// ISA fragment layouts (CDNA5 ISA):
//   A 16x4 f32 : lane L<16 -> (M=L, K=k0+0..1); lane>=16 -> (M=L-16, K=k0+2..3)
//   B 4x16 f32 : lane L<16 -> (K=k0+0..1, N=L); lane>=16 -> (K=k0+2..3, N=L-16)
// => both fragments load a float2 at row_ptr + k0 + 2*(lane>=16).
//   C 16x16 f32: diag(e,e): e<8 -> lane e, vgpr e; e>=8 -> lane e+16, vgpr e-8.
// EXEC must be all-1s across the WMMA loop: indices are clamped, not branched.
// ---------------------------------------------------------------------------
__global__ __launch_bounds__(256) void edge_sim_wmma_kernel(
    const float* __restrict__ z, const long long* __restrict__ eidx,
    const float* __restrict__ invn, const float* __restrict__ pgfi,
    float* __restrict__ out, int n_edges) {
  const int lane    = threadIdx.x & 31;
  const int n_tiles = (n_edges + 15) >> 4;
  const int tile    = blockIdx.x * 8 + (threadIdx.x >> 5);
  if (tile >= n_tiles) return;               // wave-uniform: EXEC stays all-1s

  const int base = tile * 16;
  const int e    = lane & 15;                // edge-in-tile this lane serves
  int eg = base + e;
  if (eg >= n_edges) eg = n_edges - 1;       // clamp (no divergence)

  const int src = (int)eidx[eg];
  const int dst = (int)eidx[(size_t)n_edges + eg];

  const int koff = (lane >> 4) << 1;         // 0 for lanes 0-15, 2 for 16-31
  const float* __restrict__ srow = z + (size_t)src * HIDDEN + koff;
  const float* __restrict__ drow = z + (size_t)dst * HIDDEN + koff;

  v8f c = {};
#pragma unroll
  for (int k = 0; k < HIDDEN; k += 4) {
    v2f a = *(const v2f*)(srow + k);         // global_load_b64 (L2 hit)
    v2f b = *(const v2f*)(drow + k);
    // 8 args: (neg_a, A, neg_b, B, c_mod, C, reuse_a, reuse_b)
    c = __builtin_amdgcn_wmma_f32_16x16x4_f32(
        false, a, false, b, (short)0, c, false, false);
  }

  // Extract diagonal element e from the C layout.
  int r = -1;
  if (lane < 8)        r = lane;             // diag 0..7
  else if (lane >= 24) r = lane - 24;        // diag 8..15
  float d = c[0];
#pragma unroll
  for (int i = 1; i < 8; ++i) d = (r == i) ? c[i] : d;

  if (r >= 0 && (base + e) < n_edges) {
    out[base + e] = d * invn[src] * pgfi[dst];
  }
}

// ---------------------------------------------------------------------------
extern "C" void kernel_launch(void* const* d_in, const int* in_sizes, int n_in,
                              void* d_out, int out_size, void* d_ws,
                              size_t ws_size, hipStream_t stream) {
  const float* z           = (const float*)d_in[0];
  const long long* eidx    = (const long long*)d_in[1];
  const int n_nodes        = in_sizes[0] / HIDDEN;     // 100000
  const int n_edges        = in_sizes[1] / 2;          // 1600000
  float* out               = (float*)d_out;

  float* invn = (float*)d_ws;               // n_nodes floats
  float* pgfi = invn + n_nodes;             // n_nodes floats (800 KB total)

  const int node_blocks = (n_nodes + 7) / 8;            // 8 waves/block
  node_stats_kernel<<<node_blocks, 256, 0, stream>>>(z, invn, pgfi, n_nodes);

  const int n_tiles    = (n_edges + 15) / 16;
  const int edge_blocks = (n_tiles + 7) / 8;            // 8 tiles/block
  edge_sim_wmma_kernel<<<edge_blocks, 256, 0, stream>>>(z, eidx, invn, pgfi,
                                                        out, n_edges);
}